// GraphSEncoder_86784109183557
// MI455X (gfx1250) — compile-verified
//
#include <hip/hip_runtime.h>

// CDNA5 (gfx1250) wave32 WMMA fragments for V_WMMA_F32_16X16X4_F32:
//   A 16x4 : 2 VGPRs/lane. lanes 0-15: rows 0-15, K={0,1}; lanes 16-31: same rows, K={2,3}
//   B 4x16 : 2 VGPRs/lane. lanes 0-15: col=lane, K={0,1}; lanes 16-31: col=lane-16, K={2,3}
//   C 16x16: 8 VGPRs. vgpr v, lanes 0-15: M=v, N=lane; lanes 16-31: M=v+8, N=lane-16
typedef __attribute__((ext_vector_type(2))) float v2f;
typedef __attribute__((ext_vector_type(8))) float v8f;

#define N_DIS 25000
#define N_MIR 25000
#define N_TOT 50000
#define EDGES 600000
#define EMB 128

static __device__ __forceinline__ v8f wmma_f32(v2f a, v2f b, v8f c) {
  // 8 args: (neg_a, A, neg_b, B, c_mod, C, reuse_a, reuse_b)
  return __builtin_amdgcn_wmma_f32_16x16x4_f32(false, a, false, b, (short)0, c,
                                               false, false);
}

// ---------------------------------------------------------------------------
// out[M x 128] = A[M x K] @ W[K x 128] + bias   (feature projection)
// One wave per 16-row tile; 8 accumulator tiles cover all 128 output columns.
// ---------------------------------------------------------------------------
__global__ __launch_bounds__(256) void proj_kernel(
    const float* __restrict__ A, const float* __restrict__ W,
    const float* __restrict__ bias, float* __restrict__ out, int M, int K) {
  const int lane = threadIdx.x & 31;
  const int wave = threadIdx.x >> 5;
  const int ntiles = (M + 15) >> 4;
  const int tile = blockIdx.x * 8 + wave;
  if (tile >= ntiles) return;

  const int rbase = tile << 4;
  const int l16 = lane & 15;
  const int hi = lane >> 4;   // which half of the wave
  const int khalf = hi << 1;  // K sub-offset 0 or 2

  int row = rbase + l16;
  int rowc = row < M ? row : M - 1;  // clamp loads on the ragged last tile
  const float* arow = A + (size_t)rowc * K;

  v8f c[8] = {};

  const int kfull = K & ~3;
  for (int k0 = 0; k0 < kfull; k0 += 4) {
    v2f a;
    a.x = arow[k0 + khalf];
    a.y = arow[k0 + khalf + 1];
    const float* wp = W + (size_t)(k0 + khalf) * EMB + l16;
#pragma unroll
    for (int t = 0; t < 8; ++t) {
      v2f b;
      b.x = wp[t * 16];
      b.y = wp[t * 16 + EMB];
      c[t] = wmma_f32(a, b, c[t]);
    }
  }
  if (K & 3) {  // zero-padded K tail (383 = 4*95+3, 495 = 4*123+3)
    const int ka = kfull + khalf, kb = ka + 1;
    v2f a;
    a.x = (ka < K) ? arow[ka] : 0.0f;
    a.y = (kb < K) ? arow[kb] : 0.0f;
#pragma unroll
    for (int t = 0; t < 8; ++t) {
      v2f b;
      b.x = (ka < K) ? W[(size_t)ka * EMB + t * 16 + l16] : 0.0f;
      b.y = (kb < K) ? W[(size_t)kb * EMB + t * 16 + l16] : 0.0f;
      c[t] = wmma_f32(a, b, c[t]);
    }
  }

#pragma unroll
  for (int t = 0; t < 8; ++t) {
    const float bb = bias[t * 16 + l16];
#pragma unroll
    for (int v = 0; v < 8; ++v) {
      const int r = rbase + v + hi * 8;
      if (r < M) out[(size_t)r * EMB + t * 16 + l16] = c[t][v] + bb;
    }
  }
}

// ---------------------------------------------------------------------------
// out = [x @ W_self + (agg * inv_deg) @ W_neigh + b], optional ReLU.
// Mean-normalization folded into the A-fragment load (row-uniform per lane).
// ---------------------------------------------------------------------------
__global__ __launch_bounds__(256) void sage_kernel(
    const float* __restrict__ X, const float* __restrict__ AGG,
    const float* __restrict__ invdeg, const float* __restrict__ Ws,
    const float* __restrict__ Wn, const float* __restrict__ bias,
    float* __restrict__ out, int relu) {
  const int lane = threadIdx.x & 31;
  const int wave = threadIdx.x >> 5;
  const int tile = blockIdx.x * 8 + wave;  // 50000/16 = 3125 exact tiles
  if (tile >= (N_TOT >> 4)) return;

  const int rbase = tile << 4;
  const int l16 = lane & 15;
  const int hi = lane >> 4;
  const int khalf = hi << 1;

  const int row = rbase + l16;
  const float idg = invdeg[row];
  const float* xrow = X + (size_t)row * EMB;
  const float* grow = AGG + (size_t)row * EMB;

  v8f c[8] = {};

  for (int k0 = 0; k0 < EMB; k0 += 4) {
    v2f ax, ag;
    ax.x = xrow[k0 + khalf];
    ax.y = xrow[k0 + khalf + 1];
    ag.x = grow[k0 + khalf] * idg;
    ag.y = grow[k0 + khalf + 1] * idg;
    const float* wsp = Ws + (size_t)(k0 + khalf) * EMB + l16;
    const float* wnp = Wn + (size_t)(k0 + khalf) * EMB + l16;
#pragma unroll
    for (int t = 0; t < 8; ++t) {
      v2f bs, bn;
      bs.x = wsp[t * 16];
      bs.y = wsp[t * 16 + EMB];
      bn.x = wnp[t * 16];
      bn.y = wnp[t * 16 + EMB];
      c[t] = wmma_f32(ax, bs, c[t]);
      c[t] = wmma_f32(ag, bn, c[t]);
    }
  }

#pragma unroll
  for (int t = 0; t < 8; ++t) {
    const float bb = bias[t * 16 + l16];
#pragma unroll
    for (int v = 0; v < 8; ++v) {
      float val = c[t][v] + bb;
      if (relu) val = fmaxf(val, 0.0f);
      const int r = rbase + v + hi * 8;
      out[(size_t)r * EMB + t * 16 + l16] = val;
    }
  }
}

// ---------------------------------------------------------------------------
// Edge scatter: agg[dst] += h[src], one thread per (edge, float4).
// ---------------------------------------------------------------------------
__global__ __launch_bounds__(256) void scatter_kernel(
    const float* __restrict__ H, const int* __restrict__ src,
    const int* __restrict__ dst, float* __restrict__ agg) {
  const int idx = blockIdx.x * 256 + threadIdx.x;  // EDGES * 32 threads
  const int e = idx >> 5;
  if (e >= EDGES) return;
  const int q = (idx & 31) << 2;
  const int s = src[e];
  const int d = dst[e];
  const float4 v = *reinterpret_cast<const float4*>(H + (size_t)s * EMB + q);
  float* p = agg + (size_t)d * EMB + q;
  atomicAdd(p + 0, v.x);
  atomicAdd(p + 1, v.y);
  atomicAdd(p + 2, v.z);
  atomicAdd(p + 3, v.w);
}

__global__ __launch_bounds__(256) void degree_kernel(const int* __restrict__ dst,
                                                     float* __restrict__ deg) {
  const int e = blockIdx.x * 256 + threadIdx.x;
  if (e < EDGES) atomicAdd(&deg[dst[e]], 1.0f);
}

__global__ __launch_bounds__(256) void invdeg_kernel(float* __restrict__ deg) {
  const int i = blockIdx.x * 256 + threadIdx.x;
  if (i < N_TOT) deg[i] = 1.0f / fmaxf(deg[i], 1.0f);
}

__global__ __launch_bounds__(256) void zero_kernel(float* __restrict__ p, int n) {
  const int i = blockIdx.x * 256 + threadIdx.x;
  if (i < n) p[i] = 0.0f;
}

// ---------------------------------------------------------------------------
extern "C" void kernel_launch(void* const* d_in, const int* in_sizes, int n_in,
                              void* d_out, int out_size, void* d_ws,
                              size_t ws_size, hipStream_t stream) {
  (void)in_sizes; (void)n_in; (void)out_size; (void)ws_size;

  const float* d_feat   = (const float*)d_in[0];   // [25000 x 383]
  const float* m_feat   = (const float*)d_in[1];   // [25000 x 495]
  const int*   src      = (const int*)d_in[2];     // [E]
  const int*   dst      = (const int*)d_in[3];     // [E]
  const float* W_d      = (const float*)d_in[4];   // [383 x 128]
  const float* b_d      = (const float*)d_in[5];
  const float* W_m      = (const float*)d_in[6];   // [495 x 128]
  const float* b_m      = (const float*)d_in[7];
  const float* W_self1  = (const float*)d_in[8];
  const float* W_neigh1 = (const float*)d_in[9];
  const float* b1       = (const float*)d_in[10];
  const float* W_self2  = (const float*)d_in[11];
  const float* W_neigh2 = (const float*)d_in[12];
  const float* b2       = (const float*)d_in[13];

  float* h   = (float*)d_ws;                        // [50000 x 128]
  float* h1  = h + (size_t)N_TOT * EMB;             // [50000 x 128]
  float* agg = h1 + (size_t)N_TOT * EMB;            // [50000 x 128]
  float* deg = agg + (size_t)N_TOT * EMB;           // [50000] -> inv_deg

  const int NB_NODE  = (N_TOT + 255) / 256;
  const int NB_EDGE  = (EDGES + 255) / 256;
  const int NB_AGG   = (N_TOT * EMB + 255) / 256;
  const int NB_SCAT  = (EDGES * 32 + 255) / 256;
  const int NB_PROJ  = (((N_DIS + 15) / 16) + 7) / 8;
  const int NB_SAGE  = ((N_TOT / 16) + 7) / 8;

  // in-degree -> inv_deg (DGL mean aggregation; isolated nodes -> 1)
  zero_kernel<<<NB_NODE, 256, 0, stream>>>(deg, N_TOT);
  degree_kernel<<<NB_EDGE, 256, 0, stream>>>(dst, deg);
  invdeg_kernel<<<NB_NODE, 256, 0, stream>>>(deg);

  // per-type embedding projections into concatenated h
  proj_kernel<<<NB_PROJ, 256, 0, stream>>>(d_feat, W_d, b_d, h, N_DIS, 383);
  proj_kernel<<<NB_PROJ, 256, 0, stream>>>(m_feat, W_m, b_m,
                                           h + (size_t)N_DIS * EMB, N_MIR, 495);

  // SAGE layer 1 (ReLU)
  zero_kernel<<<NB_AGG, 256, 0, stream>>>(agg, N_TOT * EMB);
  scatter_kernel<<<NB_SCAT, 256, 0, stream>>>(h, src, dst, agg);
  sage_kernel<<<NB_SAGE, 256, 0, stream>>>(h, agg, deg, W_self1, W_neigh1, b1,
                                           h1, 1);

  // SAGE layer 2 (linear) -> d_out
  zero_kernel<<<NB_AGG, 256, 0, stream>>>(agg, N_TOT * EMB);
  scatter_kernel<<<NB_SCAT, 256, 0, stream>>>(h1, src, dst, agg);
  sage_kernel<<<NB_SAGE, 256, 0, stream>>>(h1, agg, deg, W_self2, W_neigh2, b2,
                                           (float*)d_out, 0);
}